// RQBottleneck_10299331576387
// MI455X (gfx1250) — compile-verified
//
#include <hip/hip_runtime.h>
#include <hip/hip_bf16.h>

typedef float v2f __attribute__((ext_vector_type(2)));
typedef float v8f __attribute__((ext_vector_type(8)));

#define NCB   4
#define NEMB  4096
#define DDIM  256
#define NTOK  65536            // B*S = 16*4096
#define QUANT_ELEMS 16777216   // 16*4096*256
#define LOSS_OFF    16777216
#define CODES_OFF   16777217

__device__ __forceinline__ v8f wmma_f32(v2f a, v2f b, v8f c) {
  // D = A(16x4 f32) * B(4x16 f32) + C(16x16 f32)
  return __builtin_amdgcn_wmma_f32_16x16x4_f32(
      /*neg_a=*/false, a, /*neg_b=*/false, b,
      /*c_mod=*/(short)0, c, /*reuse_a=*/false, /*reuse_b=*/false);
}

// Precompute ||c_n||^2 for all 4 codebooks into workspace.
__global__ void cnorm_kernel(const float* __restrict__ cbs,
                             float* __restrict__ cnorms) {
  int i = blockIdx.x * blockDim.x + threadIdx.x;   // 0 .. 16383
  if (i >= NCB * NEMB) return;
  const float* row = cbs + (size_t)i * DDIM;
  float s = 0.f;
#pragma unroll 8
  for (int d = 0; d < DDIM; d += 4) {
    v2f p0 = *(const v2f*)(row + d);
    v2f p1 = *(const v2f*)(row + d + 2);
    s += p0.x * p0.x + p0.y * p0.y + p1.x * p1.x + p1.y * p1.y;
  }
  cnorms[i] = s;
}

// Fully fused residual-VQ: one wave owns 16 tokens, residual lives in VGPRs
// as WMMA A-fragments across all 4 stages.
__global__ __launch_bounds__(256) void rvq_kernel(
    const float* __restrict__ x, const float* __restrict__ cbs,
    const float* __restrict__ cnorms, float* __restrict__ quants,
    float* __restrict__ codes, float* __restrict__ partials) {
  __shared__ int sIdx[8][16];

  const int lane = threadIdx.x & 31;
  const int wave = threadIdx.x >> 5;
  const int m    = lane & 15;                 // token row within wave / N col
  const int koff = (lane & 16) ? 2 : 0;       // f32 WMMA K-pair offset per half
  const int tok  = blockIdx.x * 128 + wave * 16 + m;
  const float* xrow = x + (size_t)tok * DDIM + koff;

  // Residual tile in registers: a[k] = residual[tok][4k+koff .. 4k+koff+1]
  v2f a[64];
#pragma unroll
  for (int k = 0; k < 64; ++k) a[k] = *(const v2f*)(xrow + 4 * k);

  float lsum = 0.f;

  for (int st = 0; st < NCB; ++st) {
    const float* cb = cbs + (size_t)st * (NEMB * DDIM);
    const float* cn = cnorms + st * NEMB;

    float bestv[8];
    int   besti[8];
#pragma unroll
    for (int v = 0; v < 8; ++v) { bestv[v] = 3.402823466e38f; besti[v] = 0x7FFFFFFF; }

#pragma unroll 1
    for (int n0 = 0; n0 < NEMB; n0 += 16) {
      // B fragment: lane covers codebook row n0+m, same koff pattern as A
      const float* bptr = cb + (size_t)(n0 + m) * DDIM + koff;
      v8f acc0 = {0.f, 0.f, 0.f, 0.f, 0.f, 0.f, 0.f, 0.f};
      v8f acc1 = {0.f, 0.f, 0.f, 0.f, 0.f, 0.f, 0.f, 0.f};
#pragma unroll
      for (int k = 0; k < 64; k += 2) {
        v2f b0 = *(const v2f*)(bptr + 4 * k);
        v2f b1 = *(const v2f*)(bptr + 4 * k + 4);
        acc0 = wmma_f32(a[k],     b0, acc0);
        acc1 = wmma_f32(a[k + 1], b1, acc1);
      }
      const float cnv = cn[n0 + m];
      const int   n   = n0 + m;
#pragma unroll
      for (int v = 0; v < 8; ++v) {
        float val = cnv - 2.0f * (acc0[v] + acc1[v]);  // dist minus const ||r||^2
        if (val < bestv[v]) { bestv[v] = val; besti[v] = n; }
      }
    }

    // Argmin across the 16 lanes of each half (rows 0-7 in lanes 0-15,
    // rows 8-15 in lanes 16-31). Tie-break on smaller index (argmin-first).
#pragma unroll
    for (int v = 0; v < 8; ++v) {
#pragma unroll
      for (int mask = 1; mask < 16; mask <<= 1) {
        float ov = __shfl_xor(bestv[v], mask, 32);
        int   oi = __shfl_xor(besti[v], mask, 32);
        if (ov < bestv[v] || (ov == bestv[v] && oi < besti[v])) {
          bestv[v] = ov; besti[v] = oi;
        }
      }
    }

    if (lane == 0) {
#pragma unroll
      for (int r = 0; r < 8; ++r) sIdx[wave][r] = besti[r];
    }
    if (lane == 16) {
#pragma unroll
      for (int r = 0; r < 8; ++r) sIdx[wave][8 + r] = besti[r];
    }
    __syncthreads();
    const int myidx = sIdx[wave][m];
    if (lane < 16) codes[(size_t)tok * NCB + st] = (float)myidx;

    // Gather winning codeword slice, update residual, accumulate loss:
    // loss_c = sum(residual_c^2) since agg_c = x - residual_c.
    const float* qptr = cb + (size_t)myidx * DDIM + koff;
#pragma unroll
    for (int k = 0; k < 64; ++k) {
      v2f q = *(const v2f*)(qptr + 4 * k);
      a[k] -= q;
      lsum += a[k].x * a[k].x + a[k].y * a[k].y;
    }
    __syncthreads();
  }

  // quants_trunc == agg_last == x - residual_final
#pragma unroll
  for (int k = 0; k < 64; ++k) {
    v2f xv = *(const v2f*)(xrow + 4 * k);
    *(v2f*)(quants + (size_t)tok * DDIM + koff + 4 * k) = xv - a[k];
  }

  // Deterministic loss partial per wave
#pragma unroll
  for (int mask = 16; mask >= 1; mask >>= 1) lsum += __shfl_xor(lsum, mask, 32);
  if (lane == 0) partials[blockIdx.x * 8 + wave] = lsum;
}

__global__ void loss_reduce_kernel(const float* __restrict__ partials,
                                   float* __restrict__ loss_out) {
  __shared__ float sm[256];
  int t = threadIdx.x;
  float s = 0.f;
  for (int i = t; i < 4096; i += 256) s += partials[i];
  sm[t] = s;
  __syncthreads();
  for (int off = 128; off > 0; off >>= 1) {
    if (t < off) sm[t] += sm[t + off];
    __syncthreads();
  }
  if (t == 0) loss_out[0] = sm[0] * (1.0f / 67108864.0f);  // /(C*B*S*D)
}

extern "C" void kernel_launch(void* const* d_in, const int* in_sizes, int n_in,
                              void* d_out, int out_size, void* d_ws, size_t ws_size,
                              hipStream_t stream) {
  const float* x   = (const float*)d_in[0];   // [16,4096,256]
  const float* cbs = (const float*)d_in[1];   // [4,4096,256]
  float* out = (float*)d_out;

  float* ws       = (float*)d_ws;
  float* cnorms   = ws;                 // 4*4096 floats
  float* partials = ws + NCB * NEMB;    // 4096 floats (one per wave)

  cnorm_kernel<<<(NCB * NEMB + 255) / 256, 256, 0, stream>>>(cbs, cnorms);

  // 65536 tokens / (8 waves * 16 tokens) = 512 blocks
  rvq_kernel<<<NTOK / 128, 256, 0, stream>>>(x, cbs, cnorms,
                                             out,               // quants
                                             out + CODES_OFF,   // codes (as float)
                                             partials);

  loss_reduce_kernel<<<1, 256, 0, stream>>>(partials, out + LOSS_OFF);
}